// Label_Correlation_Loss_1709396984423
// MI455X (gfx1250) — compile-verified
//
#include <hip/hip_runtime.h>

#define LCL_THREADS 256
#define LCL_MAX_BLOCKS 4096

typedef float f32x4 __attribute__((ext_vector_type(4)));
typedef int   i32x4 __attribute__((ext_vector_type(4)));
typedef float v2f   __attribute__((ext_vector_type(2)));
typedef float v8f   __attribute__((ext_vector_type(8)));

// Full 32-lane sum using one V_WMMA_F32_16X16X4_F32.
// A (16x4): VGPR0 = v  -> A[m][0] = v_m (lanes 0-15), A[m][2] = v_{m+16} (lanes 16-31)
//           VGPR1 = 0  -> A[m][1] = A[m][3] = 0
// B (4x16): all ones (layout-independent).
// D[m][n] = v_m + v_{m+16}. Lane j<16 sums D rows 0..7 (=> lanes 0-7,16-23),
// lane 16+j sums rows 8..15 (=> lanes 8-15,24-31); one xor-16 shuffle adds halves.
// EXEC must be all-ones at the call site (callers guarantee full reconvergence).
__device__ __forceinline__ float wave_sum32_wmma(float v) {
    v2f a; a[0] = v;    a[1] = 0.0f;
    v2f b; b[0] = 1.0f; b[1] = 1.0f;
    v8f c = {};
    c = __builtin_amdgcn_wmma_f32_16x16x4_f32(
        /*neg_a=*/false, a, /*neg_b=*/false, b,
        /*c_mod=*/(short)0, c, /*reuse_a=*/false, /*reuse_b=*/false);
    float t = ((c[0] + c[1]) + (c[2] + c[3])) + ((c[4] + c[5]) + (c[6] + c[7]));
    t += __shfl_xor(t, 16, 32);   // wave32: combine the two row-halves
    return t;
}

// acc[0] = sum_neg exp(p), acc[1] = sum_pos exp(-p), acc[2] = n_pos  (all f64)
__global__ void lcl_init(double* __restrict__ acc) {
    if (threadIdx.x < 3) acc[threadIdx.x] = 0.0;
}

__device__ __forceinline__ void lcl_accum(float x, int y,
                                          float& sneg, float& spos, float& cpos) {
    // p = sigmoid(x) in (0,1); exp(+/-p) numerically safe.
    float p   = __builtin_amdgcn_rcpf(1.0f + __expf(-x));
    float ep  = __expf(p);
    float emp = __expf(-p);
    bool  pos = (y > 0);
    sneg += pos ? 0.0f : ep;
    spos += pos ? emp  : 0.0f;
    cpos += pos ? 1.0f : 0.0f;
}

__global__ __launch_bounds__(LCL_THREADS)
void lcl_main(const float* __restrict__ pred, const int* __restrict__ label,
              double* __restrict__ acc, int n) {
    const int nvec = n >> 2;
    const f32x4* __restrict__ p4 = (const f32x4*)pred;
    const i32x4* __restrict__ y4 = (const i32x4*)label;

    float sneg = 0.0f, spos = 0.0f, cpos = 0.0f;

    const int stride = gridDim.x * blockDim.x;
    for (int i = blockIdx.x * blockDim.x + threadIdx.x; i < nvec; i += stride) {
        // Streaming, touched-once data (268MB > 192MB L2): non-temporal B128 loads.
        f32x4 x = __builtin_nontemporal_load(p4 + i);
        i32x4 y = __builtin_nontemporal_load(y4 + i);
#pragma unroll
        for (int k = 0; k < 4; ++k)
            lcl_accum(x[k], y[k], sneg, spos, cpos);
    }

    // Scalar tail (n % 4 elements) on one thread; reconverges before reduction.
    if (blockIdx.x == 0 && threadIdx.x == 0) {
        for (int i = nvec << 2; i < n; ++i)
            lcl_accum(pred[i], label[i], sneg, spos, cpos);
    }

    // Wave-level reduction via WMMA (EXEC all-ones here: no divergence at this point).
    float wneg = wave_sum32_wmma(sneg);
    float wpos = wave_sum32_wmma(spos);
    float wcnt = wave_sum32_wmma(cpos);

    // Block-level reduction: LDS float atomics (ds_add_f32) from each wave leader.
    __shared__ float sm[3];
    if (threadIdx.x < 3) sm[threadIdx.x] = 0.0f;
    __syncthreads();
    if ((threadIdx.x & 31) == 0) {
        atomicAdd(&sm[0], wneg);
        atomicAdd(&sm[1], wpos);
        atomicAdd(&sm[2], wcnt);
    }
    __syncthreads();

    // Grid-level: f64 global atomics (exact enough; ~4096 adds per accumulator).
    if (threadIdx.x == 0) {
        atomicAdd(&acc[0], (double)sm[0]);
        atomicAdd(&acc[1], (double)sm[1]);
        atomicAdd(&acc[2], (double)sm[2]);
    }
}

__global__ void lcl_final(const double* __restrict__ acc, float* __restrict__ out, int n) {
    double sneg = acc[0];
    double spos = acc[1];
    double npos = acc[2];
    double nneg = (double)n - npos;
    double ei_pos  = spos * sneg / fmax(npos * nneg, 1.0);
    double ei_zero = sneg / fmax(nneg, 1.0);
    out[0] = (float)((npos == 0.0) ? ei_zero : ei_pos);
}

extern "C" void kernel_launch(void* const* d_in, const int* in_sizes, int n_in,
                              void* d_out, int out_size, void* d_ws, size_t ws_size,
                              hipStream_t stream) {
    const float* pred  = (const float*)d_in[0];
    const int*   label = (const int*)d_in[1];
    float*       out   = (float*)d_out;
    double*      acc   = (double*)d_ws;   // 3 doubles of scratch
    const int    n     = in_sizes[0];

    lcl_init<<<1, 32, 0, stream>>>(acc);

    int nvec   = n >> 2;
    int blocks = (nvec + LCL_THREADS - 1) / LCL_THREADS;
    if (blocks > LCL_MAX_BLOCKS) blocks = LCL_MAX_BLOCKS;
    if (blocks < 1) blocks = 1;
    lcl_main<<<blocks, LCL_THREADS, 0, stream>>>(pred, label, acc, n);

    lcl_final<<<1, 1, 0, stream>>>(acc, out, n);
}